// TopK_2637109920168
// MI455X (gfx1250) — compile-verified
//
#include <hip/hip_runtime.h>
#include <math.h>

typedef __attribute__((ext_vector_type(2))) float v2f;
typedef __attribute__((ext_vector_type(8))) float v8f;
typedef __attribute__((ext_vector_type(4))) unsigned u32x4;
typedef __attribute__((ext_vector_type(8))) int i32x8;
typedef __attribute__((ext_vector_type(4))) int i32x4;

#define NROWS 500000
#define NFEAT 256
#define TOPK  1024
#define NBINS 4096
#define MAXCAND 8192

#define BTROWS 64                    // rows per block tile (4 waves x 16)
#define NBT    ((NROWS + BTROWS - 1) / BTROWS)   // 7813
#define LROWB  1040                  // padded LDS row pitch: 1024 + 16B TDM pad
#define BUFB   (BTROWS * LROWB)      // 66560 bytes per buffer

// ws layout in 32-bit words
#define W_INVNORM 0
#define W_CANDCNT 1
#define W_THRESH  2
#define W_HIST    8
#define W_CAND    (W_HIST + NBINS)
#define W_TIDX    (W_CAND + 2*MAXCAND)
#define W_TW      (W_TIDX + TOPK)
#define W_SCORES  (W_TW + TOPK)

__device__ __forceinline__ unsigned mapf(float s) {
  unsigned b = __float_as_uint(s);
  return (b & 0x80000000u) ? ~b : (b ^ 0x80000000u);
}

// Issue one TDM 2D tile load: 256 cols x 64 rows of f32 -> LDS with 16B/row pad.
// tensor dims are tile-relative; rows beyond `rows_rem` read as zero.
__device__ __forceinline__ void tdm_load_tile(const float* gsrc, unsigned lds_off,
                                              unsigned rows_rem) {
  unsigned long long ga = (unsigned long long)(uintptr_t)gsrc;
  // D# group 0: count=1 | lds_addr | global_addr[56:0] | type=2
  unsigned g0w0 = 1u;
  unsigned g0w1 = lds_off;
  unsigned g0w2 = (unsigned)ga;
  unsigned g0w3 = ((unsigned)(ga >> 32) & 0x01FFFFFFu) | 0x80000000u;
  // D# group 1: data_size=4B, pad_enable, pad_interval=256 DW, pad_amount=4 DW
  unsigned g1w0 = (2u << 16) | (1u << 20) | (7u << 22) | (3u << 25);
  unsigned g1w1 = (NFEAT & 0xFFFFu) << 16;                      // tensor_dim0 lo16
  unsigned g1w2 = (NFEAT >> 16) | ((rows_rem & 0xFFFFu) << 16); // dim0 hi | dim1 lo
  unsigned g1w3 = (rows_rem >> 16) | ((unsigned)NFEAT << 16);   // dim1 hi | tile_dim0
  unsigned g1w4 = (unsigned)BTROWS;                             // tile_dim1 | tile_dim2=0
  unsigned g1w5 = (unsigned)NFEAT;                              // tensor_dim0_stride lo32
  unsigned g1w6 = 0u;
  unsigned g1w7 = 0u;
#define RFL(x) ((unsigned)__builtin_amdgcn_readfirstlane((int)(x)))
  u32x4 g0 = {RFL(g0w0), RFL(g0w1), RFL(g0w2), RFL(g0w3)};
  i32x8 g1 = {(int)RFL(g1w0), (int)RFL(g1w1), (int)RFL(g1w2), (int)RFL(g1w3),
              (int)RFL(g1w4), (int)RFL(g1w5), (int)RFL(g1w6), (int)RFL(g1w7)};
#undef RFL
  i32x4 z4 = {0, 0, 0, 0};
  i32x8 z8 = {0, 0, 0, 0, 0, 0, 0, 0};
  __builtin_amdgcn_tensor_load_to_lds(g0, g1, z4, z4, z8, 0);
}

// ---- kernel 0: zero hist/counters, compute 1/||scorer|| ----
__global__ __launch_bounds__(256) void k_init(const float* __restrict__ scorer,
                                              float* wsf, unsigned* wsu) {
  __shared__ float red[256];
  int t = threadIdx.x;
  for (int i = t; i < NBINS; i += 256) wsu[W_HIST + i] = 0u;
  if (t == 0) { wsu[W_CANDCNT] = 0u; wsu[W_THRESH] = 0u; }
  float v = scorer[t];
  red[t] = v * v;
  __syncthreads();
  for (int s = 128; s > 0; s >>= 1) {
    if (t < s) red[t] += red[t + s];
    __syncthreads();
  }
  if (t == 0) wsf[W_INVNORM] = 1.0f / sqrtf(red[0]);
}

// ---- kernel 1: TDM double-buffered tile load + WMMA f32 16x16x4 matvec ----
__global__ __launch_bounds__(128) void k_score(const float* __restrict__ embs,
                                               const float* __restrict__ mask,
                                               const float* __restrict__ scorer,
                                               float* wsf, unsigned* wsu) {
  extern __shared__ char ldsdyn[];           // 2 * BUFB bytes
  __shared__ float4 sc4[NFEAT / 4];
  if (threadIdx.x < NFEAT / 4)
    sc4[threadIdx.x] = ((const float4*)scorer)[threadIdx.x];

  const float invn = wsf[W_INVNORM];
  const int lane = threadIdx.x & 31;
  const int wid  = threadIdx.x >> 5;         // 4 waves
  const bool hi  = lane >= 16;
  const int r    = lane & 15;
  const unsigned ldsbase = (unsigned)(uintptr_t)ldsdyn;

  // prologue: wave 0 issues TDM for this block's first tile into buffer 0
  const int bt0 = blockIdx.x;
  if (wid == 0)
    tdm_load_tile(embs + (size_t)bt0 * BTROWS * NFEAT, ldsbase,
                  (unsigned)(NROWS - bt0 * BTROWS));
  int parity = 0;

  for (int bt = bt0; bt < NBT; bt += gridDim.x) {
    const int btn = bt + gridDim.x;
    if (wid == 0) {
      if (btn < NBT) {
        tdm_load_tile(embs + (size_t)btn * BTROWS * NFEAT,
                      ldsbase + (unsigned)((parity ^ 1) * BUFB),
                      (unsigned)(NROWS - btn * BTROWS));
        __builtin_amdgcn_s_wait_tensorcnt(1);   // current tile's TDM done (in-order)
      } else {
        __builtin_amdgcn_s_wait_tensorcnt(0);
      }
    }
    __syncthreads();

    // this wave's 16-row sub-tile, row pitch 65 float4s (conflict-free banks)
    const float4* rp = (const float4*)(ldsdyn + (size_t)parity * BUFB) +
                       (size_t)(wid * 16 + r) * (LROWB / 16);
    v8f acc = {0.f, 0.f, 0.f, 0.f, 0.f, 0.f, 0.f, 0.f};
#pragma unroll 4
    for (int c = 0; c < NFEAT / 4; ++c) {
      float4 q = rp[c];
      float4 s = sc4[c];
      v2f a, b;
      a.x = hi ? q.z : q.x;  a.y = hi ? q.w : q.y;  // A 16x4 f32 fragment
      b.x = hi ? s.z : s.x;  b.y = hi ? s.w : s.y;  // B 4x16: scorer in all cols
      acc = __builtin_amdgcn_wmma_f32_16x16x4_f32(
          false, a, false, b, (short)0, acc, false, false);
    }
    // column N=0 of D: lane 0 -> rows M=0..7 in acc[0..7], lane 16 -> M=8..15
    if (lane == 0 || lane == 16) {
      const int base = bt * BTROWS + wid * 16 + (hi ? 8 : 0);
#pragma unroll
      for (int j = 0; j < 8; ++j) {
        const int row = base + j;
        if (row < NROWS) {
          float sc = acc[j] * invn + mask[row];
          wsf[W_SCORES + row] = sc;
          atomicAdd(&wsu[W_HIST + (mapf(sc) >> 20)], 1u);
        }
      }
    }
    __syncthreads();                  // everyone done reading before TDM overwrites
    parity ^= 1;
  }
}

// ---- kernel 2: find threshold bin (suffix count >= K) ----
__global__ void k_thresh(unsigned* wsu) {
  if (threadIdx.x == 0) {
    unsigned cum = 0; unsigned T = 0;
    for (int b = NBINS - 1; b >= 0; --b) {
      cum += wsu[W_HIST + b];
      if (cum >= (unsigned)TOPK) { T = (unsigned)b; break; }
    }
    wsu[W_THRESH] = T;
  }
}

// ---- kernel 3: collect candidates with bin >= T ----
__global__ __launch_bounds__(256) void k_collect(const float* wsf, unsigned* wsu) {
  int i = blockIdx.x * 256 + threadIdx.x;
  if (i >= NROWS) return;
  unsigned T = wsu[W_THRESH];
  unsigned u = mapf(wsf[W_SCORES + i]);
  if ((u >> 20) >= T) {
    unsigned p = atomicAdd(&wsu[W_CANDCNT], 1u);
    if (p < MAXCAND) {
      wsu[W_CAND + 2 * p]     = u;
      wsu[W_CAND + 2 * p + 1] = (unsigned)i;
    }
  }
}

// ---- kernel 4: bitonic sort candidates, emit top-K (idx, tanh(score)) ----
__global__ __launch_bounds__(1024) void k_sort(unsigned* wsu, float* wsf) {
  __shared__ unsigned long long keys[MAXCAND];   // 64 KB LDS
  const int t = threadIdx.x;
  const unsigned C = min(wsu[W_CANDCNT], (unsigned)MAXCAND);
  for (int i = t; i < MAXCAND; i += 1024) {
    unsigned long long kk = 0ull;
    if (i < (int)C) {
      unsigned u   = wsu[W_CAND + 2 * i];
      unsigned idx = wsu[W_CAND + 2 * i + 1];
      kk = ((unsigned long long)u << 32) | (unsigned)(~idx);  // desc score, asc idx
    }
    keys[i] = kk;
  }
  __syncthreads();
  for (int k = 2; k <= MAXCAND; k <<= 1) {
    for (int j = k >> 1; j > 0; j >>= 1) {
      for (int i = t; i < MAXCAND; i += 1024) {
        int l = i ^ j;
        if (l > i) {
          bool up = (i & k) == 0;
          unsigned long long a = keys[i], b = keys[l];
          if (up ? (a < b) : (a > b)) { keys[i] = b; keys[l] = a; }
        }
      }
      __syncthreads();
    }
  }
  if (t < TOPK) {
    unsigned long long kk = keys[t];
    unsigned u   = (unsigned)(kk >> 32);
    unsigned idx = ~(unsigned)kk;
    unsigned bits = (u & 0x80000000u) ? (u ^ 0x80000000u) : ~u;
    float sc = __uint_as_float(bits);
    wsu[W_TIDX + t] = idx;
    wsf[W_TW + t]   = tanhf(sc);
  }
}

// ---- kernel 5: gather rows, scale, transpose -> out[F, K] ----
__global__ __launch_bounds__(256) void k_out(const float* __restrict__ embs,
                                             const unsigned* wsu, const float* wsf,
                                             float* __restrict__ out) {
  int t = blockIdx.x * 256 + threadIdx.x;
  int j = t & (TOPK - 1);
  int f = t >> 10;
  unsigned idx = wsu[W_TIDX + j];
  float w = wsf[W_TW + j];
  out[f * TOPK + j] = embs[(size_t)idx * NFEAT + f] * w;
}

extern "C" void kernel_launch(void* const* d_in, const int* in_sizes, int n_in,
                              void* d_out, int out_size, void* d_ws, size_t ws_size,
                              hipStream_t stream) {
  const float* embs   = (const float*)d_in[0];
  const float* mask   = (const float*)d_in[1];
  const float* scorer = (const float*)d_in[2];
  float*    wsf = (float*)d_ws;
  unsigned* wsu = (unsigned*)d_ws;
  float*    out = (float*)d_out;

  k_init<<<1, 256, 0, stream>>>(scorer, wsf, wsu);
  k_score<<<1024, 128, 2 * BUFB, stream>>>(embs, mask, scorer, wsf, wsu);
  k_thresh<<<1, 32, 0, stream>>>(wsu);
  k_collect<<<(NROWS + 255) / 256, 256, 0, stream>>>(wsf, wsu);
  k_sort<<<1, 1024, 0, stream>>>(wsu, wsf);
  k_out<<<(NFEAT * TOPK) / 256, 256, 0, stream>>>(embs, wsu, wsf, out);
}